// NodeClassifier_72232759984610
// MI455X (gfx1250) — compile-verified
//
#include <hip/hip_runtime.h>
#include <hip/hip_bf16.h>

typedef __attribute__((ext_vector_type(2))) float v2f;
typedef __attribute__((ext_vector_type(8))) float v8f;

#define DIN 128
#define DHID 128
#define DOUT 64

__global__ void zero_f32(float* __restrict__ p, long long n) {
  long long i = (long long)blockIdx.x * blockDim.x + threadIdx.x;
  const long long step = (long long)gridDim.x * blockDim.x;
  for (; i < n; i += step) p[i] = 0.0f;
}

__global__ void degree_kernel(const int* __restrict__ src, const int* __restrict__ dst,
                              float* __restrict__ degOut, float* __restrict__ degIn, int E) {
  int i = blockIdx.x * blockDim.x + threadIdx.x;
  const int step = gridDim.x * blockDim.x;
  for (; i < E; i += step) {
    atomicAdd(&degOut[src[i]], 1.0f);
    atomicAdd(&degIn[dst[i]], 1.0f);
  }
}

__global__ void rsqrt_clip_kernel(float* __restrict__ d, int n) {
  int i = blockIdx.x * blockDim.x + threadIdx.x;
  const int step = gridDim.x * blockDim.x;
  for (; i < n; i += step) {
    float v = d[i];
    v = v < 1.0f ? 1.0f : v;
    d[i] = rsqrtf(v);
  }
}

// Out[M,N] = (A[M,K] * scale[m]) @ W[K,N], all row-major fp32.
// One wave computes a 32x32 output block as 2x2 register-tiled 16x16 WMMAs:
// per k-step: 2 A fragments + 2 B fragments -> 4 x V_WMMA_F32_16X16X4_F32
// (each loaded fragment feeds 2 WMMAs -> half the L2 traffic of 1x1 tiling,
//  and 4 independent accumulator chains hide the WMMA RAW latency).
// A frag (16x4): lanes 0-15 -> K={0,1}, lanes 16-31 -> K={2,3}  (contiguous float2)
// B frag (4x16): VGPR0 -> K=0 (lanes 0-15) / K=2 (lanes 16-31), VGPR1 -> K=1 / K=3
// C/D (16x16):   VGPR r -> M=r (lanes 0-15) / M=r+8 (lanes 16-31), N=lane%16
template <int K, int N>
__global__ __launch_bounds__(64) void gemm_rowscale_wmma(
    const float* __restrict__ A, const float* __restrict__ scale,
    const float* __restrict__ W, float* __restrict__ Out, int M) {
  const int lane = threadIdx.x & 31;
  const int wave = threadIdx.x >> 5;
  const int tileM = blockIdx.x;                 // 32-row block
  const int nsuper = blockIdx.y * 2 + wave;     // 32-col block
  if (nsuper * 32 >= N) return;                 // wave-uniform: EXEC all-ones at WMMA

  const int mr0 = tileM * 32 + (lane & 15);
  const int mr1 = mr0 + 16;
  const int m0 = (mr0 < M) ? mr0 : (M - 1);
  const int m1 = (mr1 < M) ? mr1 : (M - 1);
  const int n0 = nsuper * 32 + (lane & 15);
  const int n1 = n0 + 16;
  const int koff = (lane >> 4) << 1;            // 0 for lanes 0-15, 2 for lanes 16-31

  const float s0 = scale[m0];
  const float s1 = scale[m1];
  const float* __restrict__ arow0 = A + (size_t)m0 * K + koff;
  const float* __restrict__ arow1 = A + (size_t)m1 * K + koff;
  const float* __restrict__ bcol0 = W + (size_t)koff * N + n0;
  const float* __restrict__ bcol1 = W + (size_t)koff * N + n1;

  v8f c00 = {}, c01 = {}, c10 = {}, c11 = {};
#pragma unroll
  for (int k = 0; k < K; k += 4) {
    v2f a0 = *reinterpret_cast<const v2f*>(arow0 + k);
    v2f a1 = *reinterpret_cast<const v2f*>(arow1 + k);
    a0 = a0 * s0;  // fuse deg_out^{-1/2} row scaling into A fragments
    a1 = a1 * s1;
    v2f b0, b1;
    b0.x = bcol0[(size_t)k * N];
    b0.y = bcol0[(size_t)(k + 1) * N];
    b1.x = bcol1[(size_t)k * N];
    b1.y = bcol1[(size_t)(k + 1) * N];
    c00 = __builtin_amdgcn_wmma_f32_16x16x4_f32(false, a0, false, b0, (short)0, c00, false, false);
    c01 = __builtin_amdgcn_wmma_f32_16x16x4_f32(false, a0, false, b1, (short)0, c01, false, false);
    c10 = __builtin_amdgcn_wmma_f32_16x16x4_f32(false, a1, false, b0, (short)0, c10, false, false);
    c11 = __builtin_amdgcn_wmma_f32_16x16x4_f32(false, a1, false, b1, (short)0, c11, false, false);
  }

  const int rb0 = tileM * 32 + ((lane >> 4) << 3);        // upper 16-row tile
  const int rb1 = rb0 + 16;                               // lower 16-row tile
#pragma unroll
  for (int r = 0; r < 8; ++r) {
    const int row0 = rb0 + r;
    const int row1 = rb1 + r;
    if (row0 < M) {
      Out[(size_t)row0 * N + n0] = c00[r];
      Out[(size_t)row0 * N + n1] = c01[r];
    }
    if (row1 < M) {
      Out[(size_t)row1 * N + n0] = c10[r];
      Out[(size_t)row1 * N + n1] = c11[r];
    }
  }
}

// agg[dst[e], :] += h[src[e], :]   (float4 gather + 4 global atomic f32 adds)
template <int D>
__global__ void scatter_kernel(const int* __restrict__ src, const int* __restrict__ dst,
                               const float* __restrict__ h, float* __restrict__ agg, int E) {
  constexpr int C = D / 4;  // float4 chunks per row (power of two)
  const long long total = (long long)E * C;
  long long i = (long long)blockIdx.x * blockDim.x + threadIdx.x;
  const long long step = (long long)gridDim.x * blockDim.x;
  for (; i < total; i += step) {
    const int e = (int)(i / C);
    const int c = (int)(i - (long long)e * C);
    const int sN = src[e];
    const int dN = dst[e];
    const float4 v = *reinterpret_cast<const float4*>(h + (size_t)sN * D + c * 4);
    float* o = agg + (size_t)dN * D + c * 4;
    atomicAdd(o + 0, v.x);
    atomicAdd(o + 1, v.y);
    atomicAdd(o + 2, v.z);
    atomicAdd(o + 3, v.w);
  }
}

// outp = act(in * invIn[node] + b[d]);  in may alias outp (each elem touched once)
template <int D, int RELU>
__global__ void bias_act_kernel(const float* in, const float* __restrict__ invIn,
                                const float* __restrict__ b, float* outp, int nNodes) {
  const long long total = (long long)nNodes * D;
  long long i = (long long)blockIdx.x * blockDim.x + threadIdx.x;
  const long long step = (long long)gridDim.x * blockDim.x;
  for (; i < total; i += step) {
    const int node = (int)(i / D);
    const int d = (int)(i - (long long)node * D);
    float v = in[i] * invIn[node] + b[d];
    if (RELU) v = v > 0.0f ? v : 0.0f;
    outp[i] = v;
  }
}

extern "C" void kernel_launch(void* const* d_in, const int* in_sizes, int n_in,
                              void* d_out, int out_size, void* d_ws, size_t ws_size,
                              hipStream_t stream) {
  (void)n_in; (void)out_size; (void)ws_size;
  const int*   src = (const int*)d_in[0];
  const int*   dst = (const int*)d_in[1];
  const float* x   = (const float*)d_in[2];
  const float* W1  = (const float*)d_in[3];
  const float* b1  = (const float*)d_in[4];
  const float* W2  = (const float*)d_in[5];
  const float* b2  = (const float*)d_in[6];
  float* out = (float*)d_out;

  const int E  = in_sizes[0];
  const int NN = in_sizes[2] / DIN;  // 50000

  // Workspace layout (floats): invOut | invIn | h[NN*DHID] | agg1[NN*DHID] | h2[NN*DOUT]
  float* w = (float*)d_ws;
  const size_t padN = ((size_t)NN + 15) & ~(size_t)15;
  float* invOut = w;
  float* invIn  = invOut + padN;
  float* h      = invIn + padN;
  float* agg1   = h + (size_t)NN * DHID;
  float* h2     = agg1 + (size_t)NN * DHID;

  // degrees -> inv sqrt
  zero_f32<<<256, 256, 0, stream>>>(invOut, (long long)(2 * padN));
  zero_f32<<<4096, 256, 0, stream>>>(agg1, (long long)NN * DHID);
  zero_f32<<<2048, 256, 0, stream>>>(out, (long long)NN * DOUT);
  degree_kernel<<<1024, 256, 0, stream>>>(src, dst, invOut, invIn, E);
  rsqrt_clip_kernel<<<256, 256, 0, stream>>>(invOut, (int)(2 * padN));

  // Layer 1: h = (x * invOut) @ W1 ; agg1 = scatter-add ; h = relu(agg1*invIn + b1)
  dim3 g1((NN + 31) / 32, DHID / 64);  // (1563, 2), 2 waves/block, 32x32 per wave
  gemm_rowscale_wmma<DIN, DHID><<<g1, 64, 0, stream>>>(x, invOut, W1, h, NN);
  scatter_kernel<DHID><<<8192, 256, 0, stream>>>(src, dst, h, agg1, E);
  bias_act_kernel<DHID, 1><<<4096, 256, 0, stream>>>(agg1, invIn, b1, h, NN);

  // Layer 2: h2 = (h * invOut) @ W2 ; out = scatter-add ; out = out*invIn + b2
  dim3 g2((NN + 31) / 32, 1);          // DOUT/64 == 1
  gemm_rowscale_wmma<DHID, DOUT><<<g2, 64, 0, stream>>>(h, invOut, W2, h2, NN);
  scatter_kernel<DOUT><<<8192, 256, 0, stream>>>(src, dst, h2, out, E);
  bias_act_kernel<DOUT, 0><<<2048, 256, 0, stream>>>(out, invIn, b2, out, NN);
}